// TensoRFVMNet_57415122813502
// MI455X (gfx1250) — compile-verified
//
#include <hip/hip_runtime.h>
#include <hip/hip_bf16.h>

typedef __attribute__((ext_vector_type(2))) float v2f;
typedef __attribute__((ext_vector_type(8))) float v8f;
// 2-float vector with 4-byte alignment: lets clang emit global_load_b64 for
// dword-aligned (not 8B-aligned) pair gathers.
typedef float v2fu __attribute__((ext_vector_type(2), aligned(4)));

namespace {

constexpr int   kNPts  = 262144;
constexpr int   kN     = 500;      // spatial grid resolution
constexpr int   kNT    = 24;       // time resolution
constexpr int   kHW    = kN * kN;  // channel slice of a spatial plane
constexpr int   kHWT   = kN * kNT; // channel slice of a time plane
constexpr float kCropR  = 1.0f;
constexpr float kCropY0 = -1.0f;
constexpr float kCropY1 = 1.0f;

// Bilinear footprint, column-pair form: two row base offsets pointing at a
// clamped column pair, row weights (bounds-zeroed) and remapped pair weights.
struct Bil {
    int   r0, r1;      // element offset of (clamped) column pair in rows y0,y1
    float wy0, wy1;    // row weights, zeroed when row out of bounds
    float wl, wh;      // pair weights, remapped for column clamping/bounds
};

// 1-D lerp footprint, pair form.
struct Ler {
    int   i;           // clamped pair base
    float wl, wh;
};

// Remap x-weights onto the clamped pair (cxb, cxb+1).
// Out-of-bounds taps receive weight 0, matching the reference's inb masking.
__device__ __forceinline__ void pair_weights(int x0, float wx1, int W,
                                             int& cxb, float& wl, float& wh) {
    const int   x1  = x0 + 1;
    const float wx0 = 1.0f - wx1;
    cxb = min(max(x0, 0), W - 2);
    wl = 0.0f;
    wh = 0.0f;
    if (cxb == x0)           wl = wx0;              // interior: pair = (x0,x1)
    else if (cxb == x1)      wl = wx1;              // x0 == -1: lo elem is x1
    if (cxb + 1 == x1)       wh = wx1;              // interior
    else if (cxb + 1 == x0)  wh = wx0;              // x0 == W-1: hi elem is x0
}

__device__ __forceinline__ Bil make_bilerp(float gx, float gy, int H, int W) {
    // px indexes the W (fastest) axis, py the H axis — matches sample_mat.
    const float px  = (gx + 1.0f) * 0.5f * (float)(W - 1);
    const float py  = (gy + 1.0f) * 0.5f * (float)(H - 1);
    const float x0f = floorf(px), y0f = floorf(py);
    const float wx1 = px - x0f,   wy1 = py - y0f;
    const int   x0  = (int)x0f,   y0  = (int)y0f;
    const int   y1  = y0 + 1;

    Bil b;
    int cxb;
    pair_weights(x0, wx1, W, cxb, b.wl, b.wh);

    const int cy0 = min(max(y0, 0), H - 1);
    const int cy1 = min(max(y1, 0), H - 1);
    b.wy0 = ((y0 >= 0) && (y0 < H)) ? (1.0f - wy1) : 0.0f;
    b.wy1 = ((y1 >= 0) && (y1 < H)) ? wy1 : 0.0f;
    b.r0  = cy0 * W + cxb;
    b.r1  = cy1 * W + cxb;
    return b;
}

__device__ __forceinline__ Ler make_lerp(float c, int n) {
    const float p   = ((c + 1.0f) * (float)n - 1.0f) * 0.5f;
    const float i0f = floorf(p);
    const float w1  = p - i0f;
    const int   i0  = (int)i0f;
    Ler l;
    pair_weights(i0, w1, n, l.i, l.wl, l.wh);
    return l;
}

__device__ __forceinline__ v2fu ld2(const float* __restrict__ p, int off) {
    return *(const v2fu*)(p + off);
}

// One bilinear plane sample: two b64 pair-gathers + weight combine.
__device__ __forceinline__ float splane(const float* __restrict__ img, int cb, const Bil& b) {
    const v2fu r0 = ld2(img, cb + b.r0);
    const v2fu r1 = ld2(img, cb + b.r1);
    return b.wy0 * (r0.x * b.wl + r0.y * b.wh)
         + b.wy1 * (r1.x * b.wl + r1.y * b.wh);
}

// One 1-D lerp sample: one b64 pair-gather.
__device__ __forceinline__ float svec(const float* __restrict__ v, int cb, const Ler& l) {
    const v2fu p = ld2(v, cb + l.i);
    return p.x * l.wl + p.y * l.wh;
}

} // namespace

// One wave = 16 points. Lane L handles point L%16; lanes 0-15 compute channels
// (4k, 4k+1), lanes 16-31 compute channels (4k+2, 4k+3) of chunk k — exactly
// the V_WMMA_F32_16X16X4_F32 A-matrix VGPR layout. 16 chained f32 WMMAs
// against a block-indicator B perform the exact 64->4 grouped reduction.
__global__ __launch_bounds__(256) void tensorf_vm_kernel(
    const float* __restrict__ pts,
    const float* __restrict__ xvec, const float* __restrict__ yvec,
    const float* __restrict__ zvec,
    const float* __restrict__ YZ, const float* __restrict__ XZ,
    const float* __restrict__ XY,
    const float* __restrict__ XT, const float* __restrict__ YT,
    const float* __restrict__ ZT,
    float* __restrict__ out)      // [0, 3*NPts): color, [3*NPts, 4*NPts): sigma
{
    const int lane    = threadIdx.x & 31;
    const int wave    = threadIdx.x >> 5;
    const int half    = lane >> 4;        // 0: channels +0,+1 ; 1: channels +2,+3
    const int ptLocal = lane & 15;
    const int pbase   = blockIdx.x * 128 + wave * 16;
    const int p       = pbase + ptLocal;

    // ---- per-point setup (duplicated in both wave halves, cheap) ----
    const float4 pt = ((const float4*)pts)[p];
    const float x = pt.x, y = pt.y, z = pt.z, t = pt.w;
    const float mask =
        ((x * x + z * z <= kCropR * kCropR) && (y >= kCropY0) && (y <= kCropY1))
            ? 1.0f : 0.0f;
    const float xn = x / kCropR;
    const float zn = z / kCropR;
    const float yn = (y - kCropY0) / (kCropY1 - kCropY0) * 2.0f - 1.0f;
    const float tn = t * 2.0f - 1.0f;

    const Ler lx = make_lerp(xn, kN);
    const Ler ly = make_lerp(yn, kN);
    const Ler lz = make_lerp(zn, kN);
    const Bil bYZ = make_bilerp(yn, zn, kN, kN);
    const Bil bXZ = make_bilerp(xn, zn, kN, kN);
    const Bil bXY = make_bilerp(xn, yn, kN, kN);
    const Bil bXT = make_bilerp(xn, tn, kN, kNT);
    const Bil bYT = make_bilerp(yn, tn, kN, kNT);
    const Bil bZT = make_bilerp(zn, tn, kN, kNT);

    // ---- gather + fuse loop, reduction folded into the WMMA accumulator ----
    v8f acc = {0.f, 0.f, 0.f, 0.f, 0.f, 0.f, 0.f, 0.f};
    const int nCol = lane & 15;   // N index of this lane's B column

    #pragma unroll 1
    for (int k = 0; k < 16; ++k) {
        const int c0 = 4 * k + 2 * half;
        v2f a;
        #pragma unroll
        for (int j = 0; j < 2; ++j) {
            const int c  = c0 + j;
            const int cp = c * kHW;
            const int ct = c * kHWT;
            const int cv = c * kN;
            float r  = splane(XY, cp, bXY) * splane(ZT, ct, bZT) * svec(xvec, cv, lx);
            r       += splane(XZ, cp, bXZ) * splane(YT, ct, bYT) * svec(yvec, cv, ly);
            r       += splane(YZ, cp, bYZ) * splane(XT, ct, bXT) * svec(zvec, cv, lz);
            a[j] = r;
        }
        // B[kl][n] = (n == k>>2) for all kl: same value in both B VGPRs, so the
        // construction is independent of which K-rows sit in which half-wave.
        const float bv = (nCol == (k >> 2)) ? 1.0f : 0.0f;
        v2f b = {bv, bv};
        acc = __builtin_amdgcn_wmma_f32_16x16x4_f32(
            /*neg_a=*/false, a, /*neg_b=*/false, b,
            /*c_mod=*/(short)0, acc, /*reuse_a=*/false, /*reuse_b=*/false);
    }

    // ---- branchless epilogue ----
    // D[m][n]: VGPR j -> rows j (lanes 0-15) / j+8 (lanes 16-31), col N = lane%16.
    // Columns 0..3 hold the 4 group sums per point. Column 0 -> sigma = exp(v)*mask,
    // columns 1..3 -> color = sigmoid(v) = rcp(1 + exp(-v)); one shared native exp
    // and a computed store index keep every step branch-free.
    if (nCol < 4) {
        const bool isSigma = (nCol == 0);
        #pragma unroll
        for (int j = 0; j < 8; ++j) {
            const int   m   = j + half * 8;          // point within this wave's 16
            const float v   = acc[j];
            const float mk  = __shfl(mask, m, 32);   // lane m holds point m's mask
            const int   pp  = pbase + m;
            const float e   = __expf(isSigma ? v : -v);
            const float res = isSigma ? (e * mk)
                                      : __builtin_amdgcn_rcpf(1.0f + e);
            const int   idx = isSigma ? (3 * kNPts + pp) : (pp * 3 + (nCol - 1));
            out[idx] = res;
        }
    }
}

extern "C" void kernel_launch(void* const* d_in, const int* in_sizes, int n_in,
                              void* d_out, int out_size, void* d_ws, size_t ws_size,
                              hipStream_t stream) {
    (void)in_sizes; (void)n_in; (void)d_ws; (void)ws_size; (void)out_size;
    const float* pts  = (const float*)d_in[0];
    // d_in[1..4]: viewdirs / iteration / embedder flags — unused by reference math
    const float* xvec = (const float*)d_in[5];
    const float* yvec = (const float*)d_in[6];
    const float* zvec = (const float*)d_in[7];
    const float* YZ   = (const float*)d_in[8];
    const float* XZ   = (const float*)d_in[9];
    const float* XY   = (const float*)d_in[10];
    const float* XT   = (const float*)d_in[11];
    const float* YT   = (const float*)d_in[12];
    const float* ZT   = (const float*)d_in[13];

    dim3 grid(kNPts / 128);   // 128 points per block (8 waves x 16 points)
    dim3 block(256);
    hipLaunchKernelGGL(tensorf_vm_kernel, grid, block, 0, stream,
                       pts, xvec, yvec, zvec, YZ, XZ, XY, XT, YT, ZT,
                       (float*)d_out);
}